// QLSTM_65481071410523
// MI455X (gfx1250) — compile-verified
//
#include <hip/hip_runtime.h>
#include <math.h>

typedef float v2f __attribute__((ext_vector_type(2)));
typedef float v8f __attribute__((ext_vector_type(8)));

#define TT 256
#define BB 1024
#define DD 64
#define NN 8
#define NCOLS 32   // 4 gates * 8 hidden

// ---------------------------------------------------------------------------
// Phase 1: preAct[row][n] = sum_d X[row][d]*W[g][j][d] + b[g][j] + phi[g][j]
// row in [0, T*B), n = g*8 + j.  One wave = one 16x16 WMMA tile, K=64 in
// 16 steps of V_WMMA_F32_16X16X4_F32 (full fp32 precision).
// ---------------------------------------------------------------------------
__global__ __launch_bounds__(128)
void qlstm_gemm_wmma(const float* __restrict__ X,
                     const float* __restrict__ Wf, const float* __restrict__ bf, const float* __restrict__ pf,
                     const float* __restrict__ Wi, const float* __restrict__ bi, const float* __restrict__ pi,
                     const float* __restrict__ Wg, const float* __restrict__ bg, const float* __restrict__ pg,
                     const float* __restrict__ Wo, const float* __restrict__ bo, const float* __restrict__ po,
                     float* __restrict__ preAct)
{
    const int lane = threadIdx.x & 31;
    const int wave = threadIdx.x >> 5;
    const int rowBase = (blockIdx.x * 2 + (wave >> 1)) * 16; // 16-row tile
    const int colBase = (wave & 1) * 16;                     // 16-col tile

    const int l15 = lane & 15;
    const int hi  = lane >> 4;            // 0: K={0,1}, 1: K={2,3} halves
    const int n   = colBase + l15;        // output column for B/C fragments
    const int gate = n >> 3;
    const int j    = n & 7;

    const float* Wsel = (gate == 0) ? Wf : (gate == 1) ? Wi : (gate == 2) ? Wg : Wo;
    const float* Wrow = Wsel + j * (DD + NN);        // row j of (8 x 72) matrix
    const float  addv = (gate == 0) ? (bf[j] + pf[j])
                      : (gate == 1) ? (bi[j] + pi[j])
                      : (gate == 2) ? (bg[j] + pg[j])
                      :               (bo[j] + po[j]);

    const float* Xrow = X + (size_t)(rowBase + l15) * DD;    // A row for this lane

    v8f c = {0.f, 0.f, 0.f, 0.f, 0.f, 0.f, 0.f, 0.f};
    #pragma unroll
    for (int k0 = 0; k0 < DD; k0 += 4) {
        const int kb = k0 + hi * 2;
        v2f a, bfrag;
        a.x     = Xrow[kb];
        a.y     = Xrow[kb + 1];
        bfrag.x = Wrow[kb];
        bfrag.y = Wrow[kb + 1];
        // D = A(16x4) * B(4x16) + C   (fp32 WMMA)
        c = __builtin_amdgcn_wmma_f32_16x16x4_f32(
                false, a, false, bfrag, (short)0, c, false, false);
    }

    const int mOff = hi * 8;   // lanes 16-31 hold rows M=8..15
    #pragma unroll
    for (int i = 0; i < 8; ++i) {
        const int row = rowBase + mOff + i;
        preAct[(size_t)row * NCOLS + n] = c[i] + addv;
    }
}

// ---------------------------------------------------------------------------
// Phase 2: sequential LSTM scan, one thread per batch element (B=1024).
// Quantum layer collapsed analytically:
//   z_i  = cos(a_i + phi_i)           (phi/bias folded into preAct)
//   e_w  = z_0*...*z_w (w>=1),  e_0 = z_1*...*z_7
// ---------------------------------------------------------------------------
__device__ __forceinline__ float sigf(float x) { return 1.0f / (1.0f + expf(-x)); }

__global__ __launch_bounds__(256)
void qlstm_scan(const float* __restrict__ preAct,
                const float* __restrict__ Wf, const float* __restrict__ Wi,
                const float* __restrict__ Wg, const float* __restrict__ Wo,
                float* __restrict__ outStack,   // (T, B, 8)
                float* __restrict__ outHx,      // (B, 8)
                float* __restrict__ outCx)      // (B, 8)
{
    __shared__ float Whs[NCOLS][NN];   // Wh[n][h] = W[gate][j][64+h]

    const int tid = threadIdx.x;
    if (tid < NCOLS * NN) {
        const int n = tid >> 3;
        const int h = tid & 7;
        const int gate = n >> 3;
        const int j    = n & 7;
        const float* Wsel = (gate == 0) ? Wf : (gate == 1) ? Wi
                          : (gate == 2) ? Wg : Wo;
        Whs[n][h] = Wsel[j * (DD + NN) + DD + h];
    }
    __syncthreads();

    const int b = blockIdx.x * blockDim.x + tid;   // batch element

    float hx[NN], cx[NN];
    #pragma unroll
    for (int h = 0; h < NN; ++h) { hx[h] = 0.f; cx[h] = 0.f; }

    for (int t = 0; t < TT; ++t) {
        const float4* pa4 = (const float4*)(preAct + ((size_t)t * BB + b) * NCOLS);
        float a[NCOLS];
        #pragma unroll
        for (int q = 0; q < 8; ++q) {
            float4 v = pa4[q];
            a[4*q+0] = v.x; a[4*q+1] = v.y; a[4*q+2] = v.z; a[4*q+3] = v.w;
        }

        // recurrent contribution + cosine
        #pragma unroll
        for (int n = 0; n < NCOLS; ++n) {
            float r = a[n];
            #pragma unroll
            for (int h = 0; h < NN; ++h) r += hx[h] * Whs[n][h];
            a[n] = cosf(r);
        }

        // prefix / suffix products per gate
        float ex[NCOLS];
        #pragma unroll
        for (int g = 0; g < 4; ++g) {
            const float* z = &a[g * 8];
            float p = z[0];
            #pragma unroll
            for (int w = 1; w < 8; ++w) { p *= z[w]; ex[g*8 + w] = p; }
            float s = z[1];
            #pragma unroll
            for (int w = 2; w < 8; ++w) s *= z[w];
            ex[g*8 + 0] = s;
        }

        // LSTM cell update
        float4 o0, o1;
        float* hv = (float*)&o0;
        #pragma unroll
        for (int h = 0; h < NN; ++h) {
            const float fg = sigf(ex[h]);
            const float ig = sigf(ex[8  + h]);
            const float gg = tanhf(ex[16 + h]);
            const float og = sigf(ex[24 + h]);
            cx[h] = fg * cx[h] + ig * gg;
            hx[h] = og * tanhf(cx[h]);
        }
        o0 = make_float4(hx[0], hx[1], hx[2], hx[3]);
        o1 = make_float4(hx[4], hx[5], hx[6], hx[7]);
        (void)hv;
        float4* dst = (float4*)(outStack + ((size_t)t * BB + b) * NN);
        dst[0] = o0;
        dst[1] = o1;
    }

    float4* dh = (float4*)(outHx + (size_t)b * NN);
    dh[0] = make_float4(hx[0], hx[1], hx[2], hx[3]);
    dh[1] = make_float4(hx[4], hx[5], hx[6], hx[7]);
    float4* dc = (float4*)(outCx + (size_t)b * NN);
    dc[0] = make_float4(cx[0], cx[1], cx[2], cx[3]);
    dc[1] = make_float4(cx[4], cx[5], cx[6], cx[7]);
}

// ---------------------------------------------------------------------------
extern "C" void kernel_launch(void* const* d_in, const int* in_sizes, int n_in,
                              void* d_out, int out_size, void* d_ws, size_t ws_size,
                              hipStream_t stream)
{
    (void)in_sizes; (void)n_in; (void)out_size; (void)ws_size;

    const float* X  = (const float*)d_in[0];
    const float* Wf = (const float*)d_in[1];
    const float* bf = (const float*)d_in[2];
    const float* pf = (const float*)d_in[3];
    const float* Wi = (const float*)d_in[4];
    const float* bi = (const float*)d_in[5];
    const float* pi = (const float*)d_in[6];
    const float* Wg = (const float*)d_in[7];
    const float* bg = (const float*)d_in[8];
    const float* pg = (const float*)d_in[9];
    const float* Wo = (const float*)d_in[10];
    const float* bo = (const float*)d_in[11];
    const float* po = (const float*)d_in[12];

    float* preAct = (float*)d_ws;             // T*B*32 floats = 32 MB scratch
    float* out    = (float*)d_out;
    float* outHx  = out + (size_t)TT * BB * NN;
    float* outCx  = outHx + (size_t)BB * NN;

    // 262144 rows / 32 rows-per-block = 8192 blocks; 4 waves/block,
    // each wave computes one 16x16 tile of the (T*B x 32) pre-activation GEMM.
    qlstm_gemm_wmma<<<dim3((TT * BB) / 32), dim3(128), 0, stream>>>(
        X, Wf, bf, pf, Wi, bi, pi, Wg, bg, pg, Wo, bo, po, preAct);

    // 1024 batch elements, fully parallel scan over T=256.
    qlstm_scan<<<dim3(BB / 256), dim3(256), 0, stream>>>(
        preAct, Wf, Wi, Wg, Wo, out, outHx, outCx);
}